// PMDElement_83047487635646
// MI455X (gfx1250) — compile-verified
//
#include <hip/hip_runtime.h>
#include <math.h>

#define SIGMA_P_F 0.01f
#define NT 256
#define NWAVES 8
#define ITEMS 32
#define TILE (NT * ITEMS)

typedef __attribute__((ext_vector_type(2))) float v2f;
typedef __attribute__((ext_vector_type(8))) float v8f;

// 2x2 complex matrix, layout: [re00, im00, re01, im01, re10, im10, re11, im11]
struct C2 { float v[8]; };

__device__ __forceinline__ C2 c2_identity() {
    C2 z;
    z.v[0] = 1.f; z.v[1] = 0.f; z.v[2] = 0.f; z.v[3] = 0.f;
    z.v[4] = 0.f; z.v[5] = 0.f; z.v[6] = 1.f; z.v[7] = 0.f;
    return z;
}

// Z = X @ Y (complex 2x2)
__device__ __forceinline__ C2 c2_mul(const C2& X, const C2& Y) {
    C2 Z;
#pragma unroll
    for (int r = 0; r < 2; ++r) {
#pragma unroll
        for (int c = 0; c < 2; ++c) {
            float x0r = X.v[(r*2+0)*2+0], x0i = X.v[(r*2+0)*2+1];
            float x1r = X.v[(r*2+1)*2+0], x1i = X.v[(r*2+1)*2+1];
            float y0r = Y.v[(0*2+c)*2+0], y0i = Y.v[(0*2+c)*2+1];
            float y1r = Y.v[(1*2+c)*2+0], y1i = Y.v[(1*2+c)*2+1];
            Z.v[(r*2+c)*2+0] = x0r*y0r - x0i*y0i + x1r*y1r - x1i*y1i;
            Z.v[(r*2+c)*2+1] = x0r*y0i + x0i*y0r + x1r*y1i + x1i*y1r;
        }
    }
    return Z;
}

// Build J_delta from one alpha triple: J = I*cos(th) - i*(a.sigma)*sin(th).
// th = 0.01*||N(0,1)^3|| <= ~0.12, so cos(th) and sin(th)/th are evaluated as
// even polynomials in t = th^2 (trunc error ~1e-12 rel, below f32 ulp).
// Avoids sqrt, sincos and the divide entirely.
__device__ __forceinline__ C2 make_J(const float* __restrict__ al3) {
    float a0 = al3[0] * SIGMA_P_F, a1 = al3[1] * SIGMA_P_F, a2 = al3[2] * SIGMA_P_F;
    float t = a0*a0 + a1*a1 + a2*a2;                       // theta^2
    // cos(th)   = 1 - t/2 + t^2/24 - t^3/720
    float c = fmaf(t, fmaf(t, fmaf(t, -1.388888889e-3f, 4.166666667e-2f), -0.5f), 1.0f);
    // sin(th)/th = 1 - t/6 + t^2/120 - t^3/5040
    float snc = fmaf(t, fmaf(t, fmaf(t, -1.984126984e-4f, 8.333333333e-3f), -1.666666667e-1f), 1.0f);
    float s0 = a0*snc, s1 = a1*snc, s2 = a2*snc;
    C2 J;
    J.v[0] = c;    J.v[1] = -s0;   // J00 = cos - i*s*a0
    J.v[2] = -s2;  J.v[3] = -s1;   // J01 = -s*a2 - i*s*a1
    J.v[4] =  s2;  J.v[5] = -s1;   // J10 =  s*a2 - i*s*a1
    J.v[6] = c;    J.v[7] =  s0;   // J11 = cos + i*s*a0
    return J;
}

// Branchless read of element (r,c) of the 4x4 real embedding [[R,-I],[I,R]]
// of a 2x2 complex matrix stored in LDS (8 floats). One ds_load + one select.
__device__ __forceinline__ float embed_get(const float* m, int r, int c) {
    int  rH = (r >> 1) & 1;                 // r >= 2
    int  cH = (c >> 1) & 1;                 // c >= 2
    int  idx = ((r & 1) * 2 + (c & 1)) * 2 + (rH ^ cH);  // +1 selects imag part
    float x = m[idx];
    return (cH & ~rH & 1) ? -x : x;         // top-right quadrant is -I
}

// Identity-matrix element for LDS padding slot: [1,0,0,0,0,0,1,0]
__device__ __forceinline__ float ident_elem(int e) {
    return (e == 0 || e == 6) ? 1.f : 0.f;
}

// Shared lane roles for the WMMA 4x(4x4) combine.
struct LaneRole {
    int wave, half, q, iblk, r4;
};
__device__ __forceinline__ LaneRole lane_role(int tid) {
    LaneRole L;
    L.wave = tid >> 5;
    int lane = tid & 31;
    L.half = lane >> 4;       // 0: lanes 0-15, 1: lanes 16-31
    L.q    = lane & 15;
    L.iblk = L.q >> 2;        // which of 4 products this quad serves
    L.r4   = L.q & 3;         // row (A) / col (B,D) within the 4x4 block
    return L;
}

// One WMMA call computing 4 independent 4x4 real products:
//   result[i] = cur[idxL(i)] @ cur[idxR(i)],  stored to nxt[idxD(i)], i = 0..3
// idxL/idxR already clamped to the identity slot (NT) for padded products.
__device__ __forceinline__ void wmma_combine(const float* cur, float* nxt,
                                             const LaneRole& L,
                                             int tL, int tR, int tD, bool storeValid) {
    const float* mL = cur + tL * 8;
    const float* mR = cur + tR * 8;
    v2f a, b;
    // A (16x4): lane q = A-row; lanes 0-15 supply K=0,1; lanes 16-31 supply K=2,3
    a.x = embed_get(mL, L.r4, 2*L.half + 0);
    a.y = embed_get(mL, L.r4, 2*L.half + 1);
    // B (4x16): lane q = B-col; vgpr v holds row K = v + 2*half
    b.x = embed_get(mR, 2*L.half + 0, L.r4);
    b.y = embed_get(mR, 2*L.half + 1, L.r4);
    v8f d = {0.f, 0.f, 0.f, 0.f, 0.f, 0.f, 0.f, 0.f};
    d = __builtin_amdgcn_wmma_f32_16x16x4_f32(false, a, false, b,
                                              (short)0, d, false, false);
    // Scalarize accumulator with CONSTANT extracts (free register renames);
    // prevents the compiler from forming a dynamic-index extract select tree.
    float d0 = d[0], d1 = d[1], d2 = d[2], d3 = d[3];
    float d4 = d[4], d5 = d[5], d6 = d[6], d7 = d[7];
    // Writeback: lane holds D rows 8*half..8*half+7. Block iblk's rows are
    // 4*iblk..4*iblk+3 -> this lane participates iff (iblk>>1)==half.
    // Need only cols 0,1 of each diagonal block ([R;I] -> complex 2x2).
    if (storeValid && (L.r4 < 2) && ((L.iblk >> 1) == L.half)) {
        bool hi = (L.iblk & 1) != 0;       // rows land in d[4..7] vs d[0..3]
        float w0 = hi ? d4 : d0;           // embed row 0 -> Re(0, r4)
        float w1 = hi ? d5 : d1;           // embed row 1 -> Re(1, r4)
        float w2 = hi ? d6 : d2;           // embed row 2 -> Im(0, r4)
        float w3 = hi ? d7 : d3;           // embed row 3 -> Im(1, r4)
        float* o = nxt + tD * 8 + 2 * L.r4;       // 8-byte aligned
        float2 p0; p0.x = w0; p0.y = w2;          // [re(0,r4), im(0,r4)]
        float2 p1; p1.x = w1; p1.y = w3;          // [re(1,r4), im(1,r4)]
        *(float2*)(o)     = p0;                   // ds_store_b64
        *(float2*)(o + 4) = p1;                   // ds_store_b64
    }
}

// Inclusive Kogge-Stone scan over NT matrices in LDS (slot NT = identity pad).
// Combine(earlier, later) = later @ earlier. Result ends in bufA.
__device__ __forceinline__ void block_scan(float* bufA, float* bufB, int tid) {
    LaneRole L = lane_role(tid);
    if (tid < 8) {                       // init identity padding slots
        bufA[NT*8 + tid] = ident_elem(tid);
        bufB[NT*8 + tid] = ident_elem(tid);
    }
    float* cur = bufA;
    float* nxt = bufB;
    __syncthreads();
    for (int s = 1; s < NT; s <<= 1) {
        if (tid < s) {
            float4* dst = (float4*)(nxt + tid*8);
            const float4* src = (const float4*)(cur + tid*8);
            dst[0] = src[0];
            dst[1] = src[1];
        }
        const int np = NT - s;                  // products this step
        const int ncalls = (np + 3) >> 2;
        for (int call = L.wave; call < ncalls; call += NWAVES) {
            int p = call*4 + L.iblk;
            bool valid = (p < np);
            int tL = valid ? (s + p) : NT;      // later operand (or identity pad)
            int tR = valid ? p       : NT;      // earlier operand
            wmma_combine(cur, nxt, L, tL, tR, s + p, valid);
        }
        __syncthreads();
        float* tmp = cur; cur = nxt; nxt = tmp;
    }
    // 8 iterations -> result back in bufA
}

// Tree reduction (product of all NT matrices) -> bufA slot 0.
__device__ __forceinline__ void block_reduce(float* bufA, float* bufB, int tid) {
    LaneRole L = lane_role(tid);
    if (tid < 8) {
        bufA[NT*8 + tid] = ident_elem(tid);
        bufB[NT*8 + tid] = ident_elem(tid);
    }
    float* cur = bufA;
    float* nxt = bufB;
    __syncthreads();
    for (int n = NT; n > 1; n >>= 1) {
        const int n2 = n >> 1;                  // products this step
        const int ncalls = (n2 + 3) >> 2;
        for (int call = L.wave; call < ncalls; call += NWAVES) {
            int p = call*4 + L.iblk;
            bool valid = (p < n2);
            int tL = valid ? (2*p + 1) : NT;    // later operand
            int tR = valid ? (2*p)     : NT;    // earlier operand
            wmma_combine(cur, nxt, L, tL, tR, p, valid);
        }
        __syncthreads();
        float* tmp = cur; cur = nxt; nxt = tmp;
    }
    // 8 iterations -> result in bufA slot 0
}

// ---------------- Kernel A: per-block aggregate ----------------
__global__ __launch_bounds__(NT)
void pmd_kA(const float* __restrict__ alphas, float* __restrict__ aggs, int T) {
    __shared__ float bufA[(NT+1)*8];
    __shared__ float bufB[(NT+1)*8];
    const int b = blockIdx.x, tid = threadIdx.x;
    const int base = b * TILE + tid * ITEMS;
    C2 P = c2_identity();
#pragma unroll 4
    for (int i = 0; i < ITEMS; ++i) {
        int k = base + i;
        if (k < T) {
            __builtin_prefetch(alphas + (k + 16) * 3, 0, 1);
            C2 J = make_J(alphas + k*3);
            P = c2_mul(J, P);
        }
    }
#pragma unroll
    for (int e = 0; e < 8; ++e) bufA[tid*8 + e] = P.v[e];
    block_reduce(bufA, bufB, tid);
    if (tid < 8) aggs[b*8 + tid] = bufA[tid];   // block product in slot 0
}

// ---------------- Kernel B: scan block aggregates (one block) ----------------
__global__ __launch_bounds__(NT)
void pmd_kB(const float* __restrict__ aggs, float* __restrict__ excl, int nb) {
    __shared__ float bufA[(NT+1)*8];
    __shared__ float bufB[(NT+1)*8];
    const int tid = threadIdx.x;
    C2 M = c2_identity();
    if (tid < nb) {
#pragma unroll
        for (int e = 0; e < 8; ++e) M.v[e] = aggs[tid*8 + e];
    }
#pragma unroll
    for (int e = 0; e < 8; ++e) bufA[tid*8 + e] = M.v[e];
    block_scan(bufA, bufB, tid);
    if (tid < nb) {
        if (tid == 0) {
#pragma unroll
            for (int e = 0; e < 8; ++e) excl[e] = ident_elem(e);
        } else {
#pragma unroll
            for (int e = 0; e < 8; ++e) excl[tid*8 + e] = bufA[(tid-1)*8 + e];
        }
    }
}

// ---------------- Kernel C: seeded scan + apply ----------------
__global__ __launch_bounds__(NT)
void pmd_kC(const float* __restrict__ sre, const float* __restrict__ sim,
            const float* __restrict__ alphas,
            const float* __restrict__ J0re, const float* __restrict__ J0im,
            const float* __restrict__ excl, float* __restrict__ out, int T) {
    __shared__ float bufA[(NT+1)*8];
    __shared__ float bufB[(NT+1)*8];
    const int b = blockIdx.x, tid = threadIdx.x;
    const int base = b * TILE + tid * ITEMS;

    // pass 1: thread-local product
    C2 P = c2_identity();
#pragma unroll 4
    for (int i = 0; i < ITEMS; ++i) {
        int k = base + i;
        if (k < T) {
            __builtin_prefetch(alphas + (k + 16) * 3, 0, 1);
            C2 J = make_J(alphas + k*3);
            P = c2_mul(J, P);
        }
    }
#pragma unroll
    for (int e = 0; e < 8; ++e) bufA[tid*8 + e] = P.v[e];
    block_scan(bufA, bufB, tid);

    // thread-exclusive in-block prefix
    C2 Epre = c2_identity();
    if (tid > 0) {
#pragma unroll
        for (int e = 0; e < 8; ++e) Epre.v[e] = bufA[(tid-1)*8 + e];
    }
    // exclusive block prefix
    C2 Bex;
#pragma unroll
    for (int e = 0; e < 8; ++e) Bex.v[e] = excl[b*8 + e];
    // initial Jones matrix J0
    C2 J0;
    J0.v[0] = J0re[0]; J0.v[1] = J0im[0]; J0.v[2] = J0re[1]; J0.v[3] = J0im[1];
    J0.v[4] = J0re[2]; J0.v[5] = J0im[2]; J0.v[6] = J0re[3]; J0.v[7] = J0im[3];

    // S = Epre @ Bex @ J0  == chain over all items before this thread, times J0
    C2 S = c2_mul(Epre, c2_mul(Bex, J0));

    // pass 2: replay items, apply to signal
#pragma unroll 4
    for (int i = 0; i < ITEMS; ++i) {
        int k = base + i;
        if (k < T) {
            __builtin_prefetch(sre + (k + 16) * 2, 0, 1);
            __builtin_prefetch(sim + (k + 16) * 2, 0, 1);
            C2 J = make_J(alphas + k*3);
            S = c2_mul(J, S);
            float2 gr = *(const float2*)(sre + 2*k);   // 8B aligned
            float2 gi = *(const float2*)(sim + 2*k);
            float4 y;
            y.x = S.v[0]*gr.x - S.v[1]*gi.x + S.v[2]*gr.y - S.v[3]*gi.y;
            y.y = S.v[0]*gi.x + S.v[1]*gr.x + S.v[2]*gi.y + S.v[3]*gr.y;
            y.z = S.v[4]*gr.x - S.v[5]*gi.x + S.v[6]*gr.y - S.v[7]*gi.y;
            y.w = S.v[4]*gi.x + S.v[5]*gr.x + S.v[6]*gi.y + S.v[7]*gr.y;
            *(float4*)(out + 4*k) = y;                 // global_store_b128
        }
    }
}

extern "C" void kernel_launch(void* const* d_in, const int* in_sizes, int n_in,
                              void* d_out, int out_size, void* d_ws, size_t ws_size,
                              hipStream_t stream) {
    const float* sre    = (const float*)d_in[0];   // signal_re [T,2]
    const float* sim    = (const float*)d_in[1];   // signal_im [T,2]
    const float* alphas = (const float*)d_in[2];   // alphas    [T,3]
    const float* J0re   = (const float*)d_in[3];   // [2,2]
    const float* J0im   = (const float*)d_in[4];   // [2,2]
    float* out = (float*)d_out;                    // complex64 [T,2] as interleaved floats

    const int T  = in_sizes[2] / 3;
    const int nb = (T + TILE - 1) / TILE;          // 245 for T=2e6 (<= NT)

    float* aggs = (float*)d_ws;                    // nb * 8 floats
    float* excl = aggs + (size_t)nb * 8;           // nb * 8 floats

    pmd_kA<<<nb, NT, 0, stream>>>(alphas, aggs, T);
    pmd_kB<<<1,  NT, 0, stream>>>(aggs, excl, nb);
    pmd_kC<<<nb, NT, 0, stream>>>(sre, sim, alphas, J0re, J0im, excl, out, T);
}